// PGDKernel_54932631715898
// MI455X (gfx1250) — compile-verified
//
#include <hip/hip_runtime.h>
#include <hip/hip_bf16.h>

// ---------------------------------------------------------------------------
// PGD kernel for MI455X (gfx1250, wave32).
// Pipeline: CIC paint -> 3D FFT (six-step 128 = 8x16 with WMMA radix-16)
//           -> spectral kernel -> per-axis gradient + inverse FFT
//           -> CIC read + position update. Everything f32, L2-resident.
// Workspace layout (floats): A_re | A_im | B_re | B_im  (4 x 128^3 = 32 MB)
// CDNA5 paths: v_wmma_f32_16x16x4_f32 for the radix-16 DFT matmul,
//              global_load_async_to_lds / global_store_async_from_lds for
//              FFT line staging (ASYNCcnt), s_wait_asynccnt sync.
// ---------------------------------------------------------------------------

#define MESH   128
#define MMASK  127
#define M2     (MESH*MESH)       // 16384
#define M3     (MESH*MESH*MESH)  // 2097152
#define NPART  M3
#define ALPHA  0.2f
#define KL2    0.09f             // KL^2, KL = 0.3
// KS = 1.0 -> KS^4 = 1

typedef float v2f __attribute__((ext_vector_type(2)));
typedef float v8f __attribute__((ext_vector_type(8)));

typedef __attribute__((address_space(1))) int* as1_i32p;
typedef __attribute__((address_space(3))) int* as3_i32p;

__device__ inline float kfreq(int i) {
    int s = (i < MESH / 2) ? i : i - MESH;
    return 6.283185307179586476925f * (float)s / (float)MESH;
}

// ---- CDNA5 async global<->LDS staging (ASYNCcnt path), with fallback ------
__device__ inline void async_g2l_b32(const float* g, float* l) {
#if __has_builtin(__builtin_amdgcn_global_load_async_to_lds_b32)
    __builtin_amdgcn_global_load_async_to_lds_b32((as1_i32p)g, (as3_i32p)l, 0, 0);
#else
    *l = *g;
#endif
}

__device__ inline void async_l2g_b32(const float* l, float* g) {
#if __has_builtin(__builtin_amdgcn_global_store_async_from_lds_b32)
    __builtin_amdgcn_global_store_async_from_lds_b32((as1_i32p)g, (as3_i32p)l, 0, 0);
#else
    *g = *l;
#endif
}

__device__ inline void wait_async_zero() {
#if __has_builtin(__builtin_amdgcn_global_load_async_to_lds_b32)
#if __has_builtin(__builtin_amdgcn_s_wait_asynccnt)
    __builtin_amdgcn_s_wait_asynccnt(0);
#else
    asm volatile("s_wait_asynccnt 0x0" ::: "memory");
#endif
#endif
}

// ---------------------------------------------------------------------------
__global__ void zero2_kernel(float* __restrict__ a, float* __restrict__ b, int n) {
    int i = blockIdx.x * blockDim.x + threadIdx.x;
    int stride = gridDim.x * blockDim.x;
    for (; i < n; i += stride) { a[i] = 0.0f; b[i] = 0.0f; }
}

// ---------------------------------------------------------------------------
__global__ void cic_paint_kernel(const float* __restrict__ pos, float* __restrict__ grid) {
    int p = blockIdx.x * blockDim.x + threadIdx.x;
    if (p >= NPART) return;
    float x = pos[p * 3 + 0], y = pos[p * 3 + 1], z = pos[p * 3 + 2];
    float fx = floorf(x), fy = floorf(y), fz = floorf(z);
    float dx = x - fx, dy = y - fy, dz = z - fz;
    int ix = ((int)fx) & MMASK, iy = ((int)fy) & MMASK, iz = ((int)fz) & MMASK;
    int xs[2] = { ix, (ix + 1) & MMASK };
    int ys[2] = { iy, (iy + 1) & MMASK };
    int zs[2] = { iz, (iz + 1) & MMASK };
    float wx[2] = { 1.0f - dx, dx };
    float wy[2] = { 1.0f - dy, dy };
    float wz[2] = { 1.0f - dz, dz };
    #pragma unroll
    for (int a = 0; a < 2; ++a)
        #pragma unroll
        for (int b = 0; b < 2; ++b)
            #pragma unroll
            for (int c = 0; c < 2; ++c)
                atomicAdd(&grid[(xs[a] * MESH + ys[b]) * MESH + zs[c]],
                          wx[a] * wy[b] * wz[c]);
}

// ---------------------------------------------------------------------------
// One FFT sweep along one axis. Each block: 16 lines of 128; each wave: 2 lines.
// 128-pt DFT = (16 x radix-8 DFT over n2) * twiddle * (DFT16 matmul via WMMA).
//   X[8*k1+k2] = sum_n1 W16^(n1 k1) * W128^(n1 k2) * sum_n2 W8^(n2 k2) x[n1+16 n2]
// sgn=+1 forward (exp(-i...)), sgn=-1 inverse; scale=1/128 per inverse pass.
// ---------------------------------------------------------------------------
#define LSTR 144   // padded line stride in floats (144 % 64 == 16 -> clean banks)
#define TSTR 17    // padded 16x16 transpose-tile row stride (conflict-free)

__launch_bounds__(256)
__global__ void fft_pass_kernel(float* __restrict__ re, float* __restrict__ im,
                                int elem_stride, int lo_count, int lo_stride,
                                int hi_stride, float sgn, float scale) {
    __shared__ float lre[16 * LSTR];
    __shared__ float lim[16 * LSTR];
    __shared__ float tbr[8 * 16 * TSTR];
    __shared__ float tbi[8 * 16 * TSTR];

    const int tid = threadIdx.x;
    const int line0 = blockIdx.x * 16;
    const int lo0 = line0 % lo_count;
    const int hi  = line0 / lo_count;
    const int base0 = hi * hi_stride + lo0 * lo_stride;
    const bool contig = (elem_stride == 1);

    // ---- stage in: async global -> LDS, 16 lines x 128 complex ------------
    #pragma unroll
    for (int i = 0; i < 8; ++i) {
        int t = tid + i * 256;
        int ln, e;
        if (contig) { ln = t >> 7; e = t & 127; }
        else        { ln = t & 15; e = t >> 4;  }
        int g = base0 + ln * lo_stride + e * elem_stride;
        async_g2l_b32(&re[g], &lre[ln * LSTR + e]);
        async_g2l_b32(&im[g], &lim[ln * LSTR + e]);
    }
    wait_async_zero();
    __syncthreads();

    const int lane = tid & 31;
    const int w    = tid >> 5;
    const int half = lane >> 4;       // 0 -> line 2w, 1 -> line 2w+1
    const int n1   = lane & 15;
    const int myline = 2 * w + half;

    // ---- stage 1: per-lane 8-point DFT over n2 of x[n1 + 16*n2] -----------
    float xr[8], xi[8];
    {
        const float* lr = &lre[myline * LSTR];
        const float* li = &lim[myline * LSTR];
        #pragma unroll
        for (int j = 0; j < 8; ++j) { xr[j] = lr[n1 + 16 * j]; xi[j] = li[n1 + 16 * j]; }
    }
    const float C7 = 0.70710678118654752440f;
    {   // DIF span-4, twiddles W8^j = exp(-sgn*i*2pi*j/8)
        float ur, ui, vr, vi, dr, di;
        ur = xr[0]; ui = xi[0]; vr = xr[4]; vi = xi[4];
        xr[0] = ur + vr; xi[0] = ui + vi; xr[4] = ur - vr; xi[4] = ui - vi;
        ur = xr[1]; ui = xi[1]; vr = xr[5]; vi = xi[5];
        dr = ur - vr; di = ui - vi;
        xr[1] = ur + vr; xi[1] = ui + vi;
        xr[5] =  C7 * dr + sgn * C7 * di;
        xi[5] =  C7 * di - sgn * C7 * dr;
        ur = xr[2]; ui = xi[2]; vr = xr[6]; vi = xi[6];
        dr = ur - vr; di = ui - vi;
        xr[2] = ur + vr; xi[2] = ui + vi;
        xr[6] =  sgn * di;
        xi[6] = -sgn * dr;
        ur = xr[3]; ui = xi[3]; vr = xr[7]; vi = xi[7];
        dr = ur - vr; di = ui - vi;
        xr[3] = ur + vr; xi[3] = ui + vi;
        xr[7] = -C7 * dr + sgn * C7 * di;
        xi[7] = -C7 * di - sgn * C7 * dr;
    }
    #pragma unroll
    for (int h = 0; h < 8; h += 4) {  // DIF span-2, twiddles 1, -sgn*i
        float ur, ui, vr, vi, dr, di;
        ur = xr[h];     ui = xi[h];     vr = xr[h + 2]; vi = xi[h + 2];
        xr[h] = ur + vr; xi[h] = ui + vi; xr[h + 2] = ur - vr; xi[h + 2] = ui - vi;
        ur = xr[h + 1]; ui = xi[h + 1]; vr = xr[h + 3]; vi = xi[h + 3];
        dr = ur - vr; di = ui - vi;
        xr[h + 1] = ur + vr; xi[h + 1] = ui + vi;
        xr[h + 3] =  sgn * di;
        xi[h + 3] = -sgn * dr;
    }
    #pragma unroll
    for (int h = 0; h < 8; h += 2) {  // span-1
        float ur = xr[h], ui = xi[h], vr = xr[h + 1], vi = xi[h + 1];
        xr[h] = ur + vr; xi[h] = ui + vi; xr[h + 1] = ur - vr; xi[h + 1] = ui - vi;
    }

    // ---- stage 2: twiddle W128^(n1*k2), transpose to WMMA B layout --------
    float* twr = &tbr[w * 16 * TSTR];
    float* twi = &tbi[w * 16 * TSTR];
    {
        const int brev[8] = { 0, 4, 2, 6, 1, 5, 3, 7 };  // DIF output order
        #pragma unroll
        for (int k2 = 0; k2 < 8; ++k2) {
            float ar = xr[brev[k2]], ai = xi[brev[k2]];
            float sn, cs;
            __sincosf(-sgn * 6.283185307179586476925f * (float)(n1 * k2) * (1.0f / 128.0f), &sn, &cs);
            twr[n1 * TSTR + 8 * half + k2] = ar * cs - ai * sn;   // B row n1, col k2+8*half
            twi[n1 * TSTR + 8 * half + k2] = ar * sn + ai * cs;
        }
    }
    __builtin_amdgcn_wave_barrier();   // same-wave LDS is in-order; block reordering

    // ---- stage 3: C(16x16) = F16(16x16) * B(16x16) complex via 16 WMMAs ---
    // A layout (16x4 f32): lane -> M=lane%16; v[0]=K(4c + 0|2), v[1]=K+1.
    // B layout mirrored; D: vgpr j -> M=j+8*(lane>=16), N=lane%16.
    v8f cr = {}; v8f ci = {};
    #pragma unroll
    for (int c = 0; c < 4; ++c) {
        int K0 = 4 * c + ((half == 0) ? 0 : 2);
        int K1 = K0 + 1;
        float s0, c0, s1, c1;
        __sincosf(-sgn * 6.283185307179586476925f * (float)(n1 * K0) * (1.0f / 16.0f), &s0, &c0);
        __sincosf(-sgn * 6.283185307179586476925f * (float)(n1 * K1) * (1.0f / 16.0f), &s1, &c1);
        v2f Ar = { c0,  c1 };    // Re F16
        v2f Ai = { s0,  s1 };    // Im F16
        v2f An = { -s0, -s1 };   // -Im F16
        v2f Br = { twr[K0 * TSTR + n1], twr[K1 * TSTR + n1] };
        v2f Bi = { twi[K0 * TSTR + n1], twi[K1 * TSTR + n1] };
        cr = __builtin_amdgcn_wmma_f32_16x16x4_f32(false, Ar, false, Br, (short)0, cr, false, false);
        cr = __builtin_amdgcn_wmma_f32_16x16x4_f32(false, An, false, Bi, (short)0, cr, false, false);
        ci = __builtin_amdgcn_wmma_f32_16x16x4_f32(false, Ar, false, Bi, (short)0, ci, false, false);
        ci = __builtin_amdgcn_wmma_f32_16x16x4_f32(false, Ai, false, Br, (short)0, ci, false, false);
    }
    __builtin_amdgcn_wave_barrier();

    // ---- scatter D back into line buffers: k = 8*k1 + k2 ------------------
    {
        int lineOut = 2 * w + (n1 >> 3);   // N = n1: cols 0-7 -> line 2w, 8-15 -> 2w+1
        int k2 = n1 & 7;
        float* orp = &lre[lineOut * LSTR];
        float* oip = &lim[lineOut * LSTR];
        #pragma unroll
        for (int j = 0; j < 8; ++j) {
            int k1 = j + 8 * half;
            orp[8 * k1 + k2] = cr[j] * scale;
            oip[8 * k1 + k2] = ci[j] * scale;
        }
    }
    __syncthreads();

    // ---- stage out: async LDS -> global -----------------------------------
    #pragma unroll
    for (int i = 0; i < 8; ++i) {
        int t = tid + i * 256;
        int ln, e;
        if (contig) { ln = t >> 7; e = t & 127; }
        else        { ln = t & 15; e = t >> 4;  }
        int g = base0 + ln * lo_stride + e * elem_stride;
        async_l2g_b32(&lre[ln * LSTR + e], &re[g]);
        async_l2g_b32(&lim[ln * LSTR + e], &im[g]);
    }
    wait_async_zero();   // (S_ENDPGM also implies wait-idle)
}

// ---------------------------------------------------------------------------
__global__ void spectral_kernel(float* __restrict__ re, float* __restrict__ im) {
    int idx = blockIdx.x * blockDim.x + threadIdx.x;
    int stride = gridDim.x * blockDim.x;
    for (; idx < M3; idx += stride) {
        int iz = idx & MMASK, iy = (idx >> 7) & MMASK, ix = idx >> 14;
        float kx = kfreq(ix), ky = kfreq(iy), kz = kfreq(iz);
        float kk = kx * kx + ky * ky + kz * kz;
        float f = 0.0f;
        if (kk != 0.0f)
            f = -(1.0f / kk) * __expf(-KL2 / kk) * __expf(-(kk * kk));  // KS^4 = 1
        re[idx] *= f;
        im[idx] *= f;
    }
}

// ---------------------------------------------------------------------------
__global__ void grad_kernel(const float* __restrict__ pre, const float* __restrict__ pim,
                            float* __restrict__ bre, float* __restrict__ bim, int axis) {
    int idx = blockIdx.x * blockDim.x + threadIdx.x;
    int stride = gridDim.x * blockDim.x;
    for (; idx < M3; idx += stride) {
        int iz = idx & MMASK, iy = (idx >> 7) & MMASK, ix = idx >> 14;
        float k = kfreq(axis == 0 ? ix : (axis == 1 ? iy : iz));
        // -i*k*(r + i*m) = k*m - i*k*r
        bre[idx] =  k * pim[idx];
        bim[idx] = -k * pre[idx];
    }
}

// ---------------------------------------------------------------------------
__global__ void cic_read_update_kernel(const float* __restrict__ grid,
                                       const float* __restrict__ pos,
                                       float* __restrict__ outpos, int axis) {
    int p = blockIdx.x * blockDim.x + threadIdx.x;
    if (p >= NPART) return;
    float x = pos[p * 3 + 0], y = pos[p * 3 + 1], z = pos[p * 3 + 2];
    float fx = floorf(x), fy = floorf(y), fz = floorf(z);
    float dx = x - fx, dy = y - fy, dz = z - fz;
    int ix = ((int)fx) & MMASK, iy = ((int)fy) & MMASK, iz = ((int)fz) & MMASK;
    int xs[2] = { ix, (ix + 1) & MMASK };
    int ys[2] = { iy, (iy + 1) & MMASK };
    int zs[2] = { iz, (iz + 1) & MMASK };
    float wx[2] = { 1.0f - dx, dx };
    float wy[2] = { 1.0f - dy, dy };
    float wz[2] = { 1.0f - dz, dz };
    float f = 0.0f;
    #pragma unroll
    for (int a = 0; a < 2; ++a)
        #pragma unroll
        for (int b = 0; b < 2; ++b)
            #pragma unroll
            for (int c = 0; c < 2; ++c)
                f += wx[a] * wy[b] * wz[c] *
                     grid[(xs[a] * MESH + ys[b]) * MESH + zs[c]];
    outpos[p * 3 + axis] = pos[p * 3 + axis] + ALPHA * f;
}

// ---------------------------------------------------------------------------
__global__ void copy_kernel(const float* __restrict__ src, float* __restrict__ dst, int n) {
    int i = blockIdx.x * blockDim.x + threadIdx.x;
    int stride = gridDim.x * blockDim.x;
    for (; i < n; i += stride) dst[i] = src[i];
}

// ---------------------------------------------------------------------------
extern "C" void kernel_launch(void* const* d_in, const int* in_sizes, int n_in,
                              void* d_out, int out_size, void* d_ws, size_t ws_size,
                              hipStream_t stream) {
    (void)in_sizes; (void)n_in; (void)out_size; (void)ws_size;
    const float* pos = (const float*)d_in[0];   // [N,3]
    const float* vel = (const float*)d_in[1];   // [N,3]
    float* out = (float*)d_out;                 // new_pos [N,3] ++ vel [N,3]

    float* Are = (float*)d_ws;                  // needs 4*M3 floats = 32 MB
    float* Aim = Are + M3;
    float* Bre = Aim + M3;
    float* Bim = Bre + M3;

    zero2_kernel<<<2048, 256, 0, stream>>>(Are, Aim, M3);
    cic_paint_kernel<<<NPART / 256, 256, 0, stream>>>(pos, Are);

    // forward FFT: axis z (stride 1), axis y (stride 128), axis x (stride 16384)
    fft_pass_kernel<<<1024, 256, 0, stream>>>(Are, Aim,     1, M2,  128,   0, 1.0f, 1.0f);
    fft_pass_kernel<<<1024, 256, 0, stream>>>(Are, Aim,  MESH, MESH,  1,  M2, 1.0f, 1.0f);
    fft_pass_kernel<<<1024, 256, 0, stream>>>(Are, Aim,    M2, MESH,  1, MESH, 1.0f, 1.0f);

    spectral_kernel<<<2048, 256, 0, stream>>>(Are, Aim);   // A = pot_k

    const float inv = 1.0f / (float)MESH;
    for (int axis = 0; axis < 3; ++axis) {
        grad_kernel<<<2048, 256, 0, stream>>>(Are, Aim, Bre, Bim, axis);
        fft_pass_kernel<<<1024, 256, 0, stream>>>(Bre, Bim,    M2, MESH,  1, MESH, -1.0f, inv);
        fft_pass_kernel<<<1024, 256, 0, stream>>>(Bre, Bim,  MESH, MESH,  1,   M2, -1.0f, inv);
        fft_pass_kernel<<<1024, 256, 0, stream>>>(Bre, Bim,     1,   M2, 128,    0, -1.0f, inv);
        cic_read_update_kernel<<<NPART / 256, 256, 0, stream>>>(Bre, pos, out, axis);
    }

    copy_kernel<<<2048, 256, 0, stream>>>(vel, out + 3 * NPART, 3 * NPART);
}